// co_HiGCN_84327387889726
// MI455X (gfx1250) — compile-verified
//
#include <hip/hip_runtime.h>

// ---------------------------------------------------------------------------
// HiGCN forward for MI455X (gfx1250, wave32).
//  - Dense GEMMs via V_WMMA_F32_16X16X4_F32, with B staged in LDS
//    (transposed, stride K+2 to dodge bank conflicts) so the inner loop is
//    global_load_b64(A) + ds_load_b64(B) + v_wmma, nothing else.
//  - K-hop propagation via CSR row-gather SpMM (atomic-free per hop;
//    whole working set ~100MB lives in the 192MB L2).
// ---------------------------------------------------------------------------

typedef float v2f __attribute__((ext_vector_type(2)));
typedef float v8f __attribute__((ext_vector_type(8)));

#define HID 64

// ---------------- WMMA GEMM: C[r,c] = act( sum_k A(r,k)*B[k,c] + bias[c] ) --
// A is split across two pointers: A(r,k) = k<K0 ? A0[r*K0+k] : A1[r*64+(k-K0)]
// handled as two sequential K segments (no per-iteration select).
// Each wave computes one 16x16 tile. Block = 256 threads = 8 waves,
// tiled as (8/colTiles) row-tiles x colTiles col-tiles.
__global__ void wmma_gemm_kernel(const float* __restrict__ A0,
                                 const float* __restrict__ A1,
                                 int K0, int K,
                                 const float* __restrict__ B, int Bcols,
                                 const float* __restrict__ bias,
                                 float* __restrict__ C, int ldc,
                                 int n, int leaky, int colTiles) {
  extern __shared__ float Bs[];  // [BsCols][K+2], transposed, zero-padded cols
  const int BsCols   = colTiles * 16;
  const int BsStride = K + 2;    // even -> v2f stays 8B aligned; kills 64-bank conflict

  // ---- cooperative fill of B into LDS (transposed) ----
  for (int i = threadIdx.x; i < BsCols * K; i += blockDim.x) {
    const int cc = i / K;        // padded column
    const int kk = i - cc * K;   // K index
    Bs[cc * BsStride + kk] =
        (cc < Bcols) ? B[(size_t)kk * Bcols + cc] : 0.0f;
  }
  __syncthreads();

  const int wave = threadIdx.x >> 5;
  const int lane = threadIdx.x & 31;
  const int colTile = wave % colTiles;
  const int rowTile = wave / colTiles;
  const int rowsPerBlock = (8 / colTiles) * 16;
  const int rowBase = blockIdx.x * rowsPerBlock + rowTile * 16;
  const int colBase = colTile * 16;

  const int half = lane >> 4;   // 0: lanes 0-15, 1: lanes 16-31
  const int idx  = lane & 15;

  // Clamp (not branch) so EXEC stays all-ones for WMMA.
  int arow = rowBase + idx; if (arow > n - 1) arow = n - 1;
  const int bcol = colBase + idx;            // global output column
  const float* BsCol = Bs + (size_t)(colBase + idx) * BsStride;

  v8f c = {};
  {
    const float* Arow = A0 + (size_t)arow * K0;
#pragma unroll 4
    for (int k = 0; k < K0; k += 4) {
      const int kk = k + 2 * half;
      v2f a = *(const v2f*)(Arow + kk);            // A[arow][kk..kk+1]
      v2f b = *(const v2f*)(BsCol + kk);           // B[kk..kk+1][bcol]
      c = __builtin_amdgcn_wmma_f32_16x16x4_f32(false, a, false, b,
                                                (short)0, c, false, false);
    }
  }
  if (K > K0) {
    const float* Arow = A1 + (size_t)arow * HID;
#pragma unroll 4
    for (int k = K0; k < K; k += 4) {
      const int kk = k + 2 * half;
      v2f a = *(const v2f*)(Arow + (kk - K0));
      v2f b = *(const v2f*)(BsCol + kk);
      c = __builtin_amdgcn_wmma_f32_16x16x4_f32(false, a, false, b,
                                                (short)0, c, false, false);
    }
  }

  const bool bvalid = (bcol < Bcols);
  const float bv = bvalid ? bias[bcol < Bcols ? bcol : 0] : 0.0f;
#pragma unroll
  for (int r = 0; r < 8; ++r) {
    const int row = rowBase + r + (half ? 8 : 0);  // documented C/D layout
    if (row < n && bvalid) {
      float v = c[r] + bv;
      if (leaky) v = (v > 0.0f) ? v : 0.01f * v;   // jax leaky_relu slope
      C[(size_t)row * ldc + bcol] = v;
    }
  }
}

// ---------------- CSR build ------------------------------------------------
__global__ void hist_kernel(const int* __restrict__ row, int* __restrict__ cnt,
                            int e) {
  int i = blockIdx.x * blockDim.x + threadIdx.x;
  if (i < e) atomicAdd(&cnt[row[i]], 1);
}

// Single-block exclusive scan of cnt[0..n-1] -> rowptr[0..n], cursor copy.
__global__ void scan_kernel(const int* __restrict__ cnt,
                            int* __restrict__ rowptr,
                            int* __restrict__ cursor, int n) {
  __shared__ int sh[1024];
  __shared__ int carry;
  const int tid = threadIdx.x;
  if (tid == 0) carry = 0;
  __syncthreads();
  for (int base = 0; base < n; base += 1024) {
    const int i = base + tid;
    const int v = (i < n) ? cnt[i] : 0;
    sh[tid] = v;
    __syncthreads();
    for (int off = 1; off < 1024; off <<= 1) {   // Hillis-Steele inclusive
      int t = (tid >= off) ? sh[tid - off] : 0;
      __syncthreads();
      sh[tid] += t;
      __syncthreads();
    }
    const int excl = sh[tid] - v + carry;
    if (i < n) { rowptr[i] = excl; cursor[i] = excl; }
    __syncthreads();
    if (tid == 0) carry += sh[1023];
    __syncthreads();
  }
  if (tid == 0) rowptr[n] = carry;
}

__global__ void scatter_kernel(const int* __restrict__ row,
                               const int* __restrict__ col,
                               const float* __restrict__ w,
                               int* __restrict__ cursor,
                               int* __restrict__ colp,
                               float* __restrict__ wp, int e) {
  int i = blockIdx.x * blockDim.x + threadIdx.x;
  if (i < e) {
    const int p = atomicAdd(&cursor[row[i]], 1);
    colp[p] = col[i];
    wp[p]   = w[i];
  }
}

// ---------------- hidden init + fused SpMM hop -----------------------------
__global__ void init_hidden_kernel(const float* __restrict__ z,
                                   const float* __restrict__ fW,
                                   float* __restrict__ h, int total) {
  int i = blockIdx.x * blockDim.x + threadIdx.x;
  if (i < total) h[i] = fW[0] * z[i];
}

// One wave per row, 2 features per lane (64 features, wave32).
// dst[r] = sum_e w_e * src[col_e];  h[r] += gamma * dst[r]  (fused, no atomics)
__global__ void spmm_hop_kernel(const int* __restrict__ rowptr,
                                const int* __restrict__ colp,
                                const float* __restrict__ wp,
                                const float* __restrict__ src,
                                float* __restrict__ dst,
                                float* __restrict__ h,
                                const float* __restrict__ fW, int hop, int n) {
  const int wave = threadIdx.x >> 5;
  const int lane = threadIdx.x & 31;
  const int r = blockIdx.x * (blockDim.x >> 5) + wave;
  if (r >= n) return;
  const float gamma = fW[hop];
  const int s = rowptr[r], e = rowptr[r + 1];
  float a0 = 0.0f, a1 = 0.0f;
  for (int i = s; i < e; ++i) {
    const int   c = colp[i];          // uniform across wave -> broadcast hit
    const float w = wp[i];
    a0 += w * src[(size_t)c * HID + lane];
    a1 += w * src[(size_t)c * HID + 32 + lane];
  }
  const size_t o = (size_t)r * HID;
  dst[o + lane]      = a0;
  dst[o + 32 + lane] = a1;
  h[o + lane]       += gamma * a0;
  h[o + 32 + lane]  += gamma * a1;
}

// ---------------------------------------------------------------------------
extern "C" void kernel_launch(void* const* d_in, const int* in_sizes, int n_in,
                              void* d_out, int out_size, void* d_ws,
                              size_t ws_size, hipStream_t stream) {
  const float* x     = (const float*)d_in[0];
  const int*   row1  = (const int*)d_in[1];
  const int*   col1  = (const int*)d_in[2];
  const float* w1    = (const float*)d_in[3];
  const int*   row2  = (const int*)d_in[4];
  const int*   col2  = (const int*)d_in[5];
  const float* w2    = (const float*)d_in[6];
  const float* fW1   = (const float*)d_in[7];
  const float* fW2   = (const float*)d_in[8];
  const float* Win1  = (const float*)d_in[9];
  const float* bin1  = (const float*)d_in[10];
  const float* Win2  = (const float*)d_in[11];
  const float* bin2  = (const float*)d_in[12];
  const float* Wout1 = (const float*)d_in[13];
  const float* bout1 = (const float*)d_in[14];
  const float* Wout2 = (const float*)d_in[15];
  const float* bout2 = (const float*)d_in[16];

  const int IN_DIM = 256, OUTD = 10;
  const int N  = in_sizes[0] / IN_DIM;
  const int E  = in_sizes[1];
  const int KH = in_sizes[7] - 1;   // number of hops

  // ---- carve workspace (all buffers fully written before any read) ----
  size_t off = 0;
  auto carve = [&](size_t bytes) -> void* {
    void* p = (char*)d_ws + off;
    off = (off + bytes + 255) & ~(size_t)255;
    return p;
  };
  float* z      = (float*)carve((size_t)N * HID * 4);
  float* h1     = (float*)carve((size_t)N * HID * 4);
  float* h2     = (float*)carve((size_t)N * HID * 4);
  float* bufA   = (float*)carve((size_t)N * HID * 4);
  float* bufB   = (float*)carve((size_t)N * HID * 4);
  int*   rowptr = (int*)carve((size_t)(N + 1) * 4);
  int*   cursor = (int*)carve((size_t)N * 4);
  int*   cnt    = (int*)carve((size_t)N * 4);
  int*   colp   = (int*)carve((size_t)E * 4);
  float* wp     = (float*)carve((size_t)E * 4);
  if (off > ws_size) return;  // workspace too small: bail deterministically

  const int TB = 256;
  const int gE  = (E + TB - 1) / TB;
  const int gNH = (N * HID + TB - 1) / TB;
  const int gGemm4 = (N + 31) / 32;    // colTiles=4 -> 32 rows/block
  const int gGemm1 = (N + 127) / 128;  // colTiles=1 -> 128 rows/block
  const int gSpmm  = (N + 7) / 8;      // 8 waves/block, one row per wave

  auto ldsB = [](int K, int colTiles) -> size_t {
    return (size_t)(colTiles * 16) * (K + 2) * sizeof(float);
  };

  auto run_branch = [&](const int* row, const int* col, const float* w,
                        const float* fW, const float* Win, const float* bin,
                        float* h) {
    // z = x @ Win + bin  (K = 256)
    wmma_gemm_kernel<<<gGemm4, TB, ldsB(IN_DIM, 4), stream>>>(
        x, x, IN_DIM, IN_DIM, Win, HID, bin, z, HID, N, 0, 4);
    // hidden = fW[0] * z
    init_hidden_kernel<<<gNH, TB, 0, stream>>>(z, fW, h, N * HID);
    // CSR build
    hipMemsetAsync(cnt, 0, (size_t)N * 4, stream);
    hist_kernel<<<gE, TB, 0, stream>>>(row, cnt, E);
    scan_kernel<<<1, 1024, 0, stream>>>(cnt, rowptr, cursor, N);
    scatter_kernel<<<gE, TB, 0, stream>>>(row, col, w, cursor, colp, wp, E);
    // K hops: xk+1 = HL * xk ; hidden += fW[k] * xk+1   (ping-pong z/A/B)
    const float* src = z;
    for (int k = 1; k <= KH; ++k) {
      float* dst = (k & 1) ? bufA : bufB;
      spmm_hop_kernel<<<gSpmm, TB, 0, stream>>>(rowptr, colp, wp, src, dst, h,
                                                fW, k, N);
      src = dst;
    }
  };

  run_branch(row1, col1, w1, fW1, Win1, bin1, h1);
  run_branch(row2, col2, w2, fW2, Win2, bin2, h2);

  // hmid = leaky_relu( cat(h1,h2) @ Wout1 + bout1 )   (K=128 via split A)
  wmma_gemm_kernel<<<gGemm4, TB, ldsB(2 * HID, 4), stream>>>(
      h1, h2, HID, 2 * HID, Wout1, HID, bout1, bufA, HID, N, 1, 4);
  // out = leaky_relu( hmid @ Wout2 + bout2 )          (Bcols=10, masked)
  wmma_gemm_kernel<<<gGemm1, TB, ldsB(HID, 1), stream>>>(
      bufA, bufA, HID, HID, Wout2, OUTD, bout2, (float*)d_out, OUTD, N, 1, 1);
}